// DictionaryLearningModule_54228257079805
// MI455X (gfx1250) — compile-verified
//
#include <hip/hip_runtime.h>
#include <math.h>

typedef __attribute__((ext_vector_type(2))) float v2f;
typedef __attribute__((ext_vector_type(8))) float v8f;

#define WMMA_F32(a, b, c) \
  __builtin_amdgcn_wmma_f32_16x16x4_f32(false, (a), false, (b), (short)0, (c), false, false)

// ---------------------------------------------------------------------------
// Small GEMM: one wave per 16x16 tile, fragments loaded straight from global
// (operands are 256KB-class, L2-resident). Compile-time variants:
//   MODE 0: C = A*B          MODE 1: C = 2*E - A*B   (Newton-Schulz step)
//   TRANSA : A stored K x M  (used for G = D^T * D)
// ---------------------------------------------------------------------------
template <int TRANSA, int MODE>
__global__ void gemm16_wave(const float* __restrict__ A, int lda,
                            const float* __restrict__ B, int ldb,
                            float* __restrict__ C, int ldc,
                            const float* __restrict__ E, int K) {
  const int lane = threadIdx.x & 31;
  const int ml = lane & 15;          // M (or N) index within tile
  const int kb = (lane >> 4) << 1;   // K sub-offset: 0 (lanes 0-15) / 2 (16-31)

  const float* pa = TRANSA ? (A + (size_t)kb * lda + blockIdx.y * 16 + ml)
                           : (A + (size_t)(blockIdx.y * 16 + ml) * lda + kb);
  const float* pb = B + (size_t)kb * ldb + blockIdx.x * 16 + ml;
  const int stepA = TRANSA ? 4 * lda : 4;
  const int stepB = 4 * ldb;

  v8f acc = {};
#pragma unroll 4
  for (int k = 0; k < K; k += 4) {
    v2f a, b;
    if (TRANSA) { a.x = pa[0]; a.y = pa[lda]; }
    else        { a = *(const v2f*)pa; }           // 8B-aligned pair
    b.x = pb[0];
    b.y = pb[ldb];
    acc = WMMA_F32(a, b, acc);
    pa += stepA;
    pb += stepB;
  }

  const int mrow = blockIdx.y * 16 + ((lane >> 4) << 3);
  const int ncol = blockIdx.x * 16 + ml;
  float* pc = C + (size_t)mrow * ldc + ncol;
  const float* pe = E + (size_t)mrow * ldc + ncol;
#pragma unroll
  for (int r = 0; r < 8; ++r) {
    float v = acc[r];
    if (MODE == 1) v = 2.0f * pe[(size_t)r * ldc] - v;
    pc[(size_t)r * ldc] = v;
  }
}

// alpha = 1 / ||G||_F   (||G||_F >= lambda_max, so rho(I - alpha*G) < 1)
__global__ void fnorm_inv_kernel(const float* __restrict__ G, float* __restrict__ alpha) {
  __shared__ float red[1024];
  const int t = threadIdx.x;
  float s = 0.0f;
  for (int i = t; i < 256 * 256; i += 1024) { float g = G[i]; s += g * g; }
  red[t] = s;
  __syncthreads();
  for (int off = 512; off > 0; off >>= 1) {
    if (t < off) red[t] += red[t + off];
    __syncthreads();
  }
  if (t == 0) *alpha = 1.0f / sqrtf(red[0]);
}

__global__ void scale_kernel(const float* __restrict__ G, const float* __restrict__ alpha,
                             float* __restrict__ X) {
  const int i = blockIdx.x * blockDim.x + threadIdx.x;
  X[i] = (*alpha) * G[i];
}

// ---------------------------------------------------------------------------
// Big GEMM: out(100000x256) = Y(100000x512) * M(512x256), f32 WMMA.
// WG = 512 threads (16 waves) -> 128x256 tile; K staged through double-
// buffered LDS in 64-wide chunks; global prefetch overlapped with WMMA.
// B is stored K-PAIRED in LDS: Bs2[p][2n] = {B[2p][n], B[2p+1][n]} so each
// WMMA B-fragment is a single aligned ds_load_b64 (no VGPR pair shuffling).
// Per wave: 16 rows x 128 cols = 8 independent v8f accumulator chains.
// ---------------------------------------------------------------------------
#define ROWS  128
#define KC    64
#define LDA_S 68     // 272B rows: 16B-aligned, bank-stride 4 -> conflict-free
#define LDB2S 544    // 2176B rows: 16B-aligned; 544%64==32 -> the two half-
                     // wave rows (p, p+1) land on disjoint 32-bank halves
#define NROW  100000

__global__ void __launch_bounds__(512)
big_gemm_kernel(const float* __restrict__ Y, const float* __restrict__ Mm,
                float* __restrict__ out) {
  __shared__ float As[2][ROWS][LDA_S];       // 2 x 34816 B
  __shared__ float Bs2[2][KC / 2][LDB2S];    // 2 x 69632 B  (total ~204 KB)

  const int tid = threadIdx.x;
  const int lane = tid & 31;
  const int wave = tid >> 5;              // 0..15
  const int m_off = (wave >> 1) << 4;     // 0,16,...,112
  const int n_off = (wave & 1) << 7;      // 0 or 128
  const int row0 = blockIdx.x * ROWS;
  const int ml = lane & 15;
  const int kb = (lane >> 4) << 1;
  const int ph = lane >> 4;               // pair-row offset for upper half-wave

  float4 ra[4];                           // staged Y tile
  float2 rb0[8], rb1[8];                  // staged M tile (row 2p / row 2p+1)

  auto loadY = [&](int kc) {
#pragma unroll
    for (int i = 0; i < 4; ++i) {
      int f = tid + i * 512;
      int r = f >> 4, c = (f & 15) << 2;
      int gr = row0 + r;
      if (gr > NROW - 1) gr = NROW - 1;   // tail-block clamp (load-safe)
      ra[i] = *(const float4*)(Y + (size_t)gr * 512 + kc + c);
    }
  };
  auto loadM = [&](int kc) {
#pragma unroll
    for (int i = 0; i < 8; ++i) {
      int g = tid + i * 512;              // 0..4095: 32 pairs x 128 n-groups
      int p = g >> 7, n2 = g & 127;
      const float* base = Mm + (size_t)(kc + 2 * p) * 256 + 2 * n2;
      rb0[i] = *(const float2*)(base);        // row 2p,   cols 2n2..2n2+1
      rb1[i] = *(const float2*)(base + 256);  // row 2p+1, cols 2n2..2n2+1
    }
  };
  auto storeAB = [&](int buf) {
#pragma unroll
    for (int i = 0; i < 4; ++i) {
      int f = tid + i * 512;
      int r = f >> 4, c = (f & 15) << 2;
      *(float4*)&As[buf][r][c] = ra[i];
    }
#pragma unroll
    for (int i = 0; i < 8; ++i) {
      int g = tid + i * 512;
      int p = g >> 7, n2 = g & 127;
      float4 v = {rb0[i].x, rb1[i].x, rb0[i].y, rb1[i].y};  // K-paired 2x2
      *(float4*)&Bs2[buf][p][n2 << 2] = v;
    }
  };

  // Prologue: stage chunk 0
  loadY(0);
  loadM(0);
  storeAB(0);
  __syncthreads();

  v8f acc[8] = {{}, {}, {}, {}, {}, {}, {}, {}};

  for (int chunk = 0; chunk < 8; ++chunk) {
    const int cur = chunk & 1;
    if (chunk < 7) {                      // issue global loads for next chunk
      loadY((chunk + 1) * KC);
      loadM((chunk + 1) * KC);
    }
    // WMMA over current LDS chunk: 16 k-steps x 8 N-tiles = 128 wmma/wave
#pragma unroll
    for (int kk = 0; kk < KC; kk += 4) {
      v2f a = *(const v2f*)&As[cur][m_off + ml][kk + kb];   // ds_load_b64
      const int p = (kk >> 1) + ph;
#pragma unroll
      for (int t = 0; t < 8; ++t) {
        v2f b = *(const v2f*)&Bs2[cur][p][(n_off + t * 16 + ml) << 1];
        acc[t] = WMMA_F32(a, b, acc[t]);
      }
    }
    if (chunk < 7) {
      storeAB(cur ^ 1);   // opposite buffer: last readers synced a chunk ago
      __syncthreads();
    }
  }

  // Epilogue: predicated stores (only the last block has partial rows)
  const int mrow = row0 + m_off + ((lane >> 4) << 3);
#pragma unroll
  for (int t = 0; t < 8; ++t) {
    const int ncol = n_off + t * 16 + ml;
#pragma unroll
    for (int r = 0; r < 8; ++r) {
      const int gr = mrow + r;
      if (gr < NROW) out[(size_t)gr * 256 + ncol] = acc[t][r];
    }
  }
}

// ---------------------------------------------------------------------------
extern "C" void kernel_launch(void* const* d_in, const int* in_sizes, int n_in,
                              void* d_out, int out_size, void* d_ws, size_t ws_size,
                              hipStream_t stream) {
  const float* Y  = (const float*)d_in[0];   // 100000 x 1 x 512
  const float* Dm = (const float*)d_in[1];   // 512 x 256
  float* out = (float*)d_out;                // 100000 x 256

  float* G     = (float*)d_ws;               // 256x256
  float* X     = G + 65536;                  // 256x256 (ping)
  float* X2    = X + 65536;                  // 256x256 (pong)
  float* T     = X2 + 65536;                 // 256x256
  float* Mmat  = T + 65536;                  // 512x256
  float* alpha = Mmat + 512 * 256;           // 1 float

  const dim3 wv(32, 1, 1);

  // G = D^T * D   (256x256, K=512)
  gemm16_wave<1, 0><<<dim3(16, 16), wv, 0, stream>>>(Dm, 256, Dm, 256, G, 256,
                                                     G, 512);
  // X0 = G / ||G||_F
  fnorm_inv_kernel<<<1, 1024, 0, stream>>>(G, alpha);
  scale_kernel<<<256, 256, 0, stream>>>(G, alpha, X);

  // Newton-Schulz: X <- 2X - X*(G*X); G SPD, kappa ~ 34 -> 16 iters >> f32 eps
  for (int it = 0; it < 16; ++it) {
    gemm16_wave<0, 0><<<dim3(16, 16), wv, 0, stream>>>(G, 256, X, 256, T, 256,
                                                       G, 256);
    gemm16_wave<0, 1><<<dim3(16, 16), wv, 0, stream>>>(X, 256, T, 256, X2, 256,
                                                       X, 256);
    float* tmp = X; X = X2; X2 = tmp;
  }

  // M = D * G^-1   (512x256, K=256)
  gemm16_wave<0, 0><<<dim3(16, 32), wv, 0, stream>>>(Dm, 256, X, 256, Mmat, 256,
                                                     Dm, 256);

  // out = Y * M    (100000x256, K=512) -- the streaming, roofline-bound GEMM
  big_gemm_kernel<<<782, 512, 0, stream>>>(Y, Mmat, out);
}